// SpecRLModel_42236708388867
// MI455X (gfx1250) — compile-verified
//
#include <hip/hip_runtime.h>

#define S_   100
#define K_   20
#define HID  256
#define EMB  64
#define TEMB 32
#define HH   128
#define NO   201
#define NR   33
#define NI   2
#define B_   512
#define A_   1024
#define ROW  384
#define OBS  8400   // S*4 + 4*S*K

typedef float v2f __attribute__((ext_vector_type(2)));
typedef float v8f __attribute__((ext_vector_type(8)));

__device__ __forceinline__ v8f wmma4(v2f a, v2f b, v8f c) {
  // V_WMMA_F32_16X16X4_F32: full fp32 16x16x4, D = A*B + C
  return __builtin_amdgcn_wmma_f32_16x16x4_f32(
      /*neg_a=*/false, a, /*neg_b=*/false, b,
      /*c_mod=*/(short)0, c, /*reuse_a=*/false, /*reuse_b=*/false);
}

// ---------------------------------------------------------------- encoder ---
struct EncParams {
  const float* obs;
  const float* emb0; const float* emb1; const float* emb2; const float* emb3;
  const float* pw;  const float* pb;
  const float* tw[4]; const float* tb[4]; const float* tg[4]; const float* tbe[4];
  const float* fw;  const float* fb; const float* fg; const float* fbe;
  float* feat;   // [B][256], pre-zeroed; accumulated mean over S
};

__global__ __launch_bounds__(256) void k_encoder(EncParams P) {
  __shared__ float sE[16 * 256];   // embeddings concat; later fusion output
  __shared__ float sRow[16 * ROW]; // fusion input rows
  __shared__ float sT[4 * 16 * K_];// clipped trace inputs
  __shared__ float sP[16 * 128];   // trace projections pre-LN

  const int b    = blockIdx.y;
  const int s0   = blockIdx.x * 16;
  const int tid  = threadIdx.x;
  const int wave = tid >> 5;
  const int lane = tid & 31;
  const int half = lane >> 4;
  const int l16  = lane & 15;

  // ---- load instruction embeddings (gather) -------------------------------
  for (int x = tid; x < 16 * 256; x += 256) {
    int r  = x >> 8;
    int c  = x & 255;
    int sg = min(s0 + r, S_ - 1);
    int fi = c >> 6;
    int j  = c & 63;
    float fv = P.obs[b * OBS + sg * 4 + fi];
    int v = (int)fv;                       // trunc toward zero, matches astype
    int V = (fi == 0) ? 202 : ((fi == 3) ? 4 : 34);
    int id = min(max(min(v, V - 1), -1) + 1, V - 1);
    const float* tab = (fi == 0) ? P.emb0 : (fi == 1) ? P.emb1
                      : (fi == 2) ? P.emb2 : P.emb3;
    sE[x] = tab[id * 64 + j];
  }
  // ---- load clipped/relu'd traces -----------------------------------------
  for (int x = tid; x < 4 * 16 * K_; x += 256) {
    int tr = x / (16 * K_);
    int rm = x % (16 * K_);
    int r  = rm / K_;
    int k  = rm % K_;
    int sg = min(s0 + r, S_ - 1);
    float v  = P.obs[b * OBS + S_ * 4 + tr * (S_ * K_) + sg * K_ + k];
    float hi = (tr < 2) ? 1e9f : 1e6f;
    sT[x] = fmaxf(fminf(v, hi), 0.0f);
  }
  __syncthreads();

  // ---- instr projection GEMM: (16x256)@(256x256), fp32 WMMA ---------------
  for (int q = 0; q < 2; ++q) {
    int n0 = (wave * 2 + q) * 16;
    v8f acc = {};
    for (int kk = 0; kk < 256; kk += 4) {
      v2f av, bv;
      av.x = sE[l16 * 256 + kk + 2 * half + 0];
      av.y = sE[l16 * 256 + kk + 2 * half + 1];
      bv.x = P.pw[(kk + 2 * half + 0) * HID + n0 + l16];
      bv.y = P.pw[(kk + 2 * half + 1) * HID + n0 + l16];
      acc = wmma4(av, bv, acc);
    }
    for (int v = 0; v < 8; ++v) {
      int r = v + 8 * half;
      sRow[r * ROW + n0 + l16] = acc[v] + P.pb[n0 + l16];
    }
  }

  // ---- trace projections (tiny: VALU) -------------------------------------
  for (int x = tid; x < 16 * 4 * TEMB; x += 256) {
    int r  = x >> 7;
    int tr = (x >> 5) & 3;
    int c  = x & 31;
    const float* w = P.tw[tr];
    float acc = P.tb[tr][c];
    const float* t = &sT[tr * 16 * K_ + r * K_];
    #pragma unroll
    for (int k = 0; k < K_; ++k) acc += t[k] * w[k * TEMB + c];
    sP[r * 128 + tr * 32 + c] = acc;
  }
  __syncthreads();
  // ---- trace LayerNorm(32) + relu -> row columns 256.. --------------------
  if (tid < 64) {
    int r = tid >> 2, tr = tid & 3;
    float* pp = &sP[r * 128 + tr * 32];
    float m = 0.f;
    for (int c = 0; c < 32; ++c) m += pp[c];
    m *= (1.0f / 32.0f);
    float vv = 0.f;
    for (int c = 0; c < 32; ++c) { float d = pp[c] - m; vv += d * d; }
    float inv = rsqrtf(vv * (1.0f / 32.0f) + 1e-5f);
    for (int c = 0; c < 32; ++c) {
      float val = (pp[c] - m) * inv * P.tg[tr][c] + P.tbe[tr][c];
      sRow[r * ROW + HID + tr * 32 + c] = fmaxf(val, 0.f);
    }
  }
  __syncthreads();

  // ---- fusion GEMM: (16x384)@(384x256), fp32 WMMA -> sE -------------------
  for (int q = 0; q < 2; ++q) {
    int n0 = (wave * 2 + q) * 16;
    v8f acc = {};
    for (int kk = 0; kk < ROW; kk += 4) {
      v2f av, bv;
      av.x = sRow[l16 * ROW + kk + 2 * half + 0];
      av.y = sRow[l16 * ROW + kk + 2 * half + 1];
      bv.x = P.fw[(kk + 2 * half + 0) * HID + n0 + l16];
      bv.y = P.fw[(kk + 2 * half + 1) * HID + n0 + l16];
      acc = wmma4(av, bv, acc);
    }
    for (int v = 0; v < 8; ++v) {
      int r = v + 8 * half;
      sE[r * HID + n0 + l16] = acc[v] + P.fb[n0 + l16];
    }
  }
  __syncthreads();

  // ---- fusion LayerNorm(256) + relu + mean-accumulate ---------------------
  for (int rr = 0; rr < 2; ++rr) {
    int r = wave * 2 + rr;
    float sum = 0.f;
    for (int c = lane; c < HID; c += 32) sum += sE[r * HID + c];
    for (int off = 16; off > 0; off >>= 1) sum += __shfl_xor(sum, off, 32);
    float m = sum * (1.0f / HID);
    float vs = 0.f;
    for (int c = lane; c < HID; c += 32) { float d = sE[r * HID + c] - m; vs += d * d; }
    for (int off = 16; off > 0; off >>= 1) vs += __shfl_xor(vs, off, 32);
    float inv = rsqrtf(vs * (1.0f / HID) + 1e-5f);
    if (s0 + r < S_) {
      for (int c = lane; c < HID; c += 32) {
        float val = fmaxf((sE[r * HID + c] - m) * inv * P.fg[c] + P.fbe[c], 0.f);
        atomicAdd(&P.feat[b * HID + c], val * (1.0f / S_));
      }
    }
  }
}

// ---------------------------------------------------------- op-head + lsm ---
__global__ __launch_bounds__(256) void k_ophead(
    const float* __restrict__ feat, const float* __restrict__ w1,
    const float* __restrict__ b1, const float* __restrict__ w2,
    const float* __restrict__ b2, float* __restrict__ op_lp) {
  __shared__ float sf[HID];
  __shared__ float sh[HH];
  __shared__ float sl[NO];
  __shared__ float red[8];
  int b = blockIdx.x, tid = threadIdx.x;
  sf[tid] = feat[b * HID + tid];
  __syncthreads();
  if (tid < HH) {
    float acc = b1[tid];
    for (int k = 0; k < HID; ++k) acc += sf[k] * w1[k * HH + tid];
    sh[tid] = fmaxf(acc, 0.f);
  }
  __syncthreads();
  if (tid < NO) {
    float acc = b2[tid];
    for (int k = 0; k < HH; ++k) acc += sh[k] * w2[k * NO + tid];
    sl[tid] = acc;
  }
  __syncthreads();
  float v = (tid < NO) ? sl[tid] : -1e30f;
  for (int off = 16; off > 0; off >>= 1) v = fmaxf(v, __shfl_xor(v, off, 32));
  if ((tid & 31) == 0) red[tid >> 5] = v;
  __syncthreads();
  float mx = red[0];
  for (int i = 1; i < 8; ++i) mx = fmaxf(mx, red[i]);
  float e = (tid < NO) ? expf(sl[tid] - mx) : 0.f;
  for (int off = 16; off > 0; off >>= 1) e += __shfl_xor(e, off, 32);
  __syncthreads();
  if ((tid & 31) == 0) red[tid >> 5] = e;
  __syncthreads();
  float s = 0.f;
  for (int i = 0; i < 8; ++i) s += red[i];
  float lse = mx + logf(s);
  if (tid < NO) op_lp[b * NO + tid] = sl[tid] - lse;
}

// -------------------------------------------- per-action embedding precomp --
struct ActParams {
  const int* at; const float* op_emb; const float* reg_emb;
  const float* rs_we;  const float* rs_b1;
  const float* rd_weo; const float* rd_wes; const float* rd_b1;
  const float* i_weo;  const float* i_wes;  const float* i_wed; const float* i_b1;
  float* ea;   // [3][A][128]
};
__global__ __launch_bounds__(128) void k_actemb(ActParams P) {
  int a = blockIdx.x, j = threadIdx.x;
  __shared__ float so[EMB], ss[EMB], sd[EMB];
  int o  = min(max(P.at[a * 4 + 0], 0), NO - 1);
  int rs = min(max(P.at[a * 4 + 1], 0), NR - 1);
  int rd = min(max(P.at[a * 4 + 2], 0), NR - 1);
  if (j < EMB) {
    so[j] = P.op_emb[o * EMB + j];
    ss[j] = P.reg_emb[rs * EMB + j];
    sd[j] = P.reg_emb[rd * EMB + j];
  }
  __syncthreads();
  float a1 = P.rs_b1[j], a2 = P.rd_b1[j], a3 = P.i_b1[j];
  for (int k = 0; k < EMB; ++k) {
    float eo = so[k], es = ss[k], ed = sd[k];
    a1 += eo * P.rs_we[k * HH + j];
    a2 += eo * P.rd_weo[k * HH + j] + es * P.rd_wes[k * HH + j];
    a3 += eo * P.i_weo[k * HH + j] + es * P.i_wes[k * HH + j] + ed * P.i_wed[k * HH + j];
  }
  P.ea[(0 * A_ + a) * HH + j] = a1;
  P.ea[(1 * A_ + a) * HH + j] = a2;
  P.ea[(2 * A_ + a) * HH + j] = a3;
}

// -------------------------------------------------------- feat projections --
__global__ __launch_bounds__(128) void k_featproj(
    const float* __restrict__ feat, const float* __restrict__ rs_wf,
    const float* __restrict__ rd_wf, const float* __restrict__ imm_wf,
    float* __restrict__ fwf /*[3][B][128]*/) {
  int b = blockIdx.x, j = threadIdx.x;
  __shared__ float sf[HID];
  sf[j] = feat[b * HID + j];
  sf[j + 128] = feat[b * HID + j + 128];
  __syncthreads();
  float a1 = 0.f, a2 = 0.f, a3 = 0.f;
  for (int k = 0; k < HID; ++k) {
    float f = sf[k];
    a1 += f * rs_wf[k * HH + j];
    a2 += f * rd_wf[k * HH + j];
    a3 += f * imm_wf[k * HH + j];
  }
  fwf[(0 * B_ + b) * HH + j] = a1;
  fwf[(1 * B_ + b) * HH + j] = a2;
  fwf[(2 * B_ + b) * HH + j] = a3;
}

// -------------------------------------------------------------- pair heads --
struct HeadParams {
  const float* fwf;   // [3][B][128]
  const float* ea;    // [3][A][128]  (biases folded)
  const float* w2_rs;  const float* b2_rs;
  const float* w2_rd;  const float* b2_rd;
  const float* w2_imm; const float* b2_imm;
  const float* op_lp;  // [B][NO]
  const int* at;
  float* out;          // [B][A]
};

__device__ __forceinline__ v8f tileGemm128(const float* sH, const float* sW,
                                           int m0, int n0, int l16, int half) {
  v8f acc = {};
  #pragma unroll
  for (int kk = 0; kk < HH; kk += 4) {
    v2f av, bv;
    av.x = sH[(m0 + l16) * HH + kk + 2 * half + 0];
    av.y = sH[(m0 + l16) * HH + kk + 2 * half + 1];
    bv.x = sW[(kk + 2 * half + 0) * 48 + n0 + l16];
    bv.y = sW[(kk + 2 * half + 1) * 48 + n0 + l16];
    acc = wmma4(av, bv, acc);
  }
  return acc;
}

__global__ __launch_bounds__(256) void k_heads(HeadParams P) {
  __shared__ float sH[64 * HH];    // h rows (relu(fwf+ea))
  __shared__ float sWL[128 * 48];  // w2 (padded) during GEMM; logits after
  __shared__ float sFwf[HH];
  __shared__ float sLp[64];

  const int b    = blockIdx.y;
  const int a0   = blockIdx.x * 64;
  const int tid  = threadIdx.x;
  const int wave = tid >> 5;
  const int lane = tid & 31;
  const int half = lane >> 4;
  const int l16  = lane & 15;

  if (tid < 64) {
    int a = a0 + tid;
    int o = min(max(P.at[a * 4 + 0], 0), NO - 1);
    sLp[tid] = P.op_lp[b * NO + o];
  }

  for (int head = 0; head < 3; ++head) {
    const float* ea  = P.ea + head * A_ * HH;
    const float* fwf = P.fwf + (head * B_ + b) * HH;
    const float* w2  = (head == 0) ? P.w2_rs : (head == 1) ? P.w2_rd : P.w2_imm;
    const float* b2  = (head == 0) ? P.b2_rs : (head == 1) ? P.b2_rd : P.b2_imm;
    const int NOUT   = (head < 2) ? NR : NI;
    const int NT     = (head < 2) ? 3 : 1;      // 16-wide N tiles

    if (tid < HH) sFwf[tid] = fwf[tid];
    __syncthreads();  // also fences reuse of sH/sWL from previous head

    for (int x = tid; x < 64 * HH; x += 256) {
      int r = x >> 7, c = x & 127;
      sH[x] = fmaxf(sFwf[c] + ea[(a0 + r) * HH + c], 0.f);
    }
    for (int x = tid; x < 128 * 48; x += 256) {
      int k = x / 48, n = x % 48;
      sWL[x] = (n < NOUT) ? w2[k * NOUT + n] : 0.f;
    }
    __syncthreads();

    const int nTiles = 4 * NT;          // M tiles = 4 (64 rows)
    int t0 = wave, t1 = wave + 8;
    bool h0 = t0 < nTiles, h1 = t1 < nTiles;
    v8f d0 = {}, d1 = {};
    if (h0) d0 = tileGemm128(sH, sWL, (t0 / NT) * 16, (t0 % NT) * 16, l16, half);
    if (h1) d1 = tileGemm128(sH, sWL, (t1 / NT) * 16, (t1 % NT) * 16, l16, half);
    __syncthreads();  // all reads of sWL (weights) done

    if (h0) {
      int m0 = (t0 / NT) * 16, n0 = (t0 % NT) * 16;
      #pragma unroll
      for (int v = 0; v < 8; ++v) sWL[(m0 + v + 8 * half) * 48 + n0 + l16] = d0[v];
    }
    if (h1) {
      int m0 = (t1 / NT) * 16, n0 = (t1 % NT) * 16;
      #pragma unroll
      for (int v = 0; v < 8; ++v) sWL[(m0 + v + 8 * half) * 48 + n0 + l16] = d1[v];
    }
    __syncthreads();

    if (tid < 64) {
      int a = a0 + tid;
      int col = (head == 0) ? min(max(P.at[a * 4 + 1], 0), NR - 1)
              : (head == 1) ? min(max(P.at[a * 4 + 2], 0), NR - 1)
                            : min(max(P.at[a * 4 + 3], 0), NI - 1);
      const float* row = &sWL[tid * 48];
      float mx = -1e30f;
      for (int n = 0; n < NOUT; ++n) mx = fmaxf(mx, row[n] + b2[n]);
      float se = 0.f;
      for (int n = 0; n < NOUT; ++n) se += expf(row[n] + b2[n] - mx);
      sLp[tid] += (row[col] + b2[col]) - (mx + logf(se));
    }
    __syncthreads();
  }
  if (tid < 64) P.out[b * A_ + a0 + tid] = sLp[tid];
}

// ------------------------------------------------------------------- host ---
extern "C" void kernel_launch(void* const* d_in, const int* in_sizes, int n_in,
                              void* d_out, int out_size, void* d_ws, size_t ws_size,
                              hipStream_t stream) {
  (void)in_sizes; (void)n_in; (void)out_size; (void)ws_size;
  const float* F[51];
  for (int i = 0; i < 51; ++i) F[i] = (const float*)d_in[i];
  const int* at = (const int*)d_in[51];

  float* ws     = (float*)d_ws;
  float* feat   = ws;                       // 512*256
  float* op_lp  = feat + B_ * HID;          // 512*201
  float* fwf    = op_lp + B_ * NO;          // 3*512*128
  float* ea     = fwf + 3 * B_ * HH;        // 3*1024*128

  hipMemsetAsync(feat, 0, (size_t)B_ * HID * sizeof(float), stream);

  EncParams ep;
  ep.obs = F[0];
  ep.emb0 = F[1]; ep.emb1 = F[2]; ep.emb2 = F[3]; ep.emb3 = F[4];
  ep.pw = F[5]; ep.pb = F[6];
  for (int t = 0; t < 4; ++t) {
    ep.tw[t]  = F[7 + t * 4 + 0];
    ep.tb[t]  = F[7 + t * 4 + 1];
    ep.tg[t]  = F[7 + t * 4 + 2];
    ep.tbe[t] = F[7 + t * 4 + 3];
  }
  ep.fw = F[23]; ep.fb = F[24]; ep.fg = F[25]; ep.fbe = F[26];
  ep.feat = feat;
  k_encoder<<<dim3(7, B_), 256, 0, stream>>>(ep);

  k_ophead<<<dim3(B_), 256, 0, stream>>>(feat, F[29], F[30], F[31], F[32], op_lp);

  ActParams ap;
  ap.at = at; ap.op_emb = F[27]; ap.reg_emb = F[28];
  ap.rs_we = F[34]; ap.rs_b1 = F[35];
  ap.rd_weo = F[39]; ap.rd_wes = F[40]; ap.rd_b1 = F[41];
  ap.i_weo = F[45]; ap.i_wes = F[46]; ap.i_wed = F[47]; ap.i_b1 = F[48];
  ap.ea = ea;
  k_actemb<<<dim3(A_), 128, 0, stream>>>(ap);

  k_featproj<<<dim3(B_), 128, 0, stream>>>(feat, F[33], F[38], F[44], fwf);

  HeadParams hp;
  hp.fwf = fwf; hp.ea = ea;
  hp.w2_rs = F[36];  hp.b2_rs = F[37];
  hp.w2_rd = F[42];  hp.b2_rd = F[43];
  hp.w2_imm = F[49]; hp.b2_imm = F[50];
  hp.op_lp = op_lp; hp.at = at; hp.out = (float*)d_out;
  k_heads<<<dim3(A_ / 64, B_), 256, 0, stream>>>(hp);
}